// JumpingKnowledge_63539746177597
// MI455X (gfx1250) — compile-verified
//
#include <hip/hip_runtime.h>

typedef __attribute__((ext_vector_type(2))) float v2f;
typedef __attribute__((ext_vector_type(8))) float v8f;

// exact-fp32 CDNA5 WMMA: D(16x16) = A(16x4) * B(4x16) + C
#define WMMA_F32(a, b, c) \
  __builtin_amdgcn_wmma_f32_16x16x4_f32(false, (a), false, (b), (short)0, (c), false, false)

// ---------------------------------------------------------------------------
// degree / normalization kernels
// ---------------------------------------------------------------------------
__global__ void fill_deg_kernel(float* __restrict__ deg, int n) {
  int i = blockIdx.x * blockDim.x + threadIdx.x;
  if (i < n) deg[i] = 1.0f;  // self-loop contributes 1
}

__global__ void count_deg_kernel(const int* __restrict__ dst, float* __restrict__ deg, int e) {
  int i = blockIdx.x * blockDim.x + threadIdx.x;
  if (i < e) atomicAdd(&deg[dst[i]], 1.0f);
}

__global__ void rsqrt_kernel(float* __restrict__ deg, int n) {
  int i = blockIdx.x * blockDim.x + threadIdx.x;
  if (i < n) deg[i] = rsqrtf(deg[i]);  // deg >= 1 always (self-loop)
}

// ---------------------------------------------------------------------------
// WMMA GEMM: out(N x 128) = sum_seg X_seg(N x 128) @ W[seg*128:(seg+1)*128, :]
// block = 128 threads (4 waves), one 16-row M tile per block,
// each wave owns two 16-col N tiles. fp32 WMMA 16x16x4.
// ---------------------------------------------------------------------------
template <int NSEG, bool RELU, bool BIAS>
__global__ __launch_bounds__(128) void gemm_wmma_kernel(
    const float* __restrict__ x0, const float* __restrict__ x1,
    const float* __restrict__ x2, const float* __restrict__ x3,
    const float* __restrict__ W,     // (NSEG*128) x 128 row-major
    const float* __restrict__ bias,  // 128 (optional)
    float* __restrict__ out)         // N x 128
{
  __shared__ __align__(16) float sX[16][132];    // X tile, row-major, padded
  __shared__ __align__(16) float sWT[128][34];   // W chunk transposed: sWT[n][k]

  const int t    = threadIdx.x;
  const int wv   = t >> 5;
  const int lane = t & 31;
  const int half = lane >> 4;   // K half select (ISA A/B layout)
  const int lrow = lane & 15;   // M for A, N for B
  const int m0   = blockIdx.x * 16;
  const int c0   = (wv * 2) * 16;
  const int c1   = c0 + 16;

  v8f acc0 = {0.f, 0.f, 0.f, 0.f, 0.f, 0.f, 0.f, 0.f};
  v8f acc1 = acc0;

  for (int seg = 0; seg < NSEG; ++seg) {
    const float* X = (seg == 0) ? x0 : (seg == 1) ? x1 : (seg == 2) ? x2 : x3;
    // stage 16x128 X tile (coalesced)
#pragma unroll
    for (int i = 0; i < 16; ++i) {
      int idx = i * 128 + t;
      int r = idx >> 7, c = idx & 127;
      sX[r][c] = X[(size_t)(m0 + r) * 128 + c];
    }
    const float* Wseg = W + (size_t)seg * 128 * 128;
    for (int kc = 0; kc < 128; kc += 32) {
      __syncthreads();  // protect sX/sWT readers from previous chunk
      // stage transposed 32-row W chunk (coalesced reads across t)
#pragma unroll
      for (int kk = 0; kk < 32; ++kk)
        sWT[t][kk] = Wseg[(size_t)(kc + kk) * 128 + t];
      __syncthreads();
#pragma unroll
      for (int k0 = 0; k0 < 32; k0 += 4) {
        const int kk = k0 + 2 * half;
        v2f a  = *(const v2f*)&sX[lrow][kc + kk];
        v2f b0 = *(const v2f*)&sWT[c0 + lrow][kk];
        v2f b1 = *(const v2f*)&sWT[c1 + lrow][kk];
        acc0 = WMMA_F32(a, b0, acc0);
        acc1 = WMMA_F32(a, b1, acc1);
      }
    }
    if (seg + 1 < NSEG) __syncthreads();  // before restaging sX
  }

  // epilogue: C/D layout -> row = j + 8*half, col = lrow (+tile offset)
  const int ccol0 = c0 + lrow;
  const int ccol1 = c1 + lrow;
  float bb0 = 0.f, bb1 = 0.f;
  if (BIAS) { bb0 = bias[ccol0]; bb1 = bias[ccol1]; }
#pragma unroll
  for (int j = 0; j < 8; ++j) {
    const int r = m0 + j + 8 * half;
    float v0 = acc0[j] + bb0;
    float v1 = acc1[j] + bb1;
    if (RELU) { v0 = fmaxf(v0, 0.f); v1 = fmaxf(v1, 0.f); }
    out[(size_t)r * 128 + ccol0] = v0;
    out[(size_t)r * 128 + ccol1] = v1;
  }
}

// ---------------------------------------------------------------------------
// aggregation kernels
// ---------------------------------------------------------------------------
__global__ __launch_bounds__(256) void self_init_kernel(
    const float* __restrict__ H, const float* __restrict__ dis,
    float* __restrict__ AGG, int n)
{
  long i = (long)blockIdx.x * blockDim.x + threadIdx.x;  // n*32 threads
  int node = (int)(i >> 5);
  if (node >= n) return;
  int l4 = (int)(i & 31);
  float w = dis[node];
  w *= w;  // dis[i]*dis[i] self-loop norm
  float4 h = ((const float4*)(H + (size_t)node * 128))[l4];
  float4 r = make_float4(h.x * w, h.y * w, h.z * w, h.w * w);
  ((float4*)(AGG + (size_t)node * 128))[l4] = r;
}

__global__ __launch_bounds__(256) void edge_agg_kernel(
    const float* __restrict__ H, const int* __restrict__ src,
    const int* __restrict__ dst, const float* __restrict__ dis,
    float* __restrict__ AGG, int e)
{
  int eid = blockIdx.x * (blockDim.x >> 5) + (threadIdx.x >> 5);
  if (eid >= e) return;
  int lane = threadIdx.x & 31;
  int s = src[eid], d = dst[eid];
  float norm = dis[s] * dis[d];
  float4 h = ((const float4*)(H + (size_t)s * 128))[lane];
  float* o = AGG + (size_t)d * 128 + lane * 4;
  atomicAdd(o + 0, h.x * norm);
  atomicAdd(o + 1, h.y * norm);
  atomicAdd(o + 2, h.z * norm);
  atomicAdd(o + 3, h.w * norm);
}

__global__ __launch_bounds__(256) void bias_relu_kernel(
    const float* __restrict__ A, const float* __restrict__ b,
    float* __restrict__ o, int n)
{
  long i = (long)blockIdx.x * blockDim.x + threadIdx.x;  // n*32 threads
  int node = (int)(i >> 5);
  if (node >= n) return;
  int l4 = (int)(i & 31);
  float4 v  = ((const float4*)(A + (size_t)node * 128))[l4];
  float4 bb = ((const float4*)b)[l4];
  float4 r = make_float4(fmaxf(v.x + bb.x, 0.f), fmaxf(v.y + bb.y, 0.f),
                         fmaxf(v.z + bb.z, 0.f), fmaxf(v.w + bb.w, 0.f));
  ((float4*)(o + (size_t)node * 128))[l4] = r;
}

// ---------------------------------------------------------------------------
// final MLP2 (128 -> 16) + row softmax; one wave per 16-row tile.
// WMMA gives lane%16 = column, j + 8*(lane/16) = row -> softmax reduces
// across the 16-lane group via shfl_xor butterflies.
// ---------------------------------------------------------------------------
__global__ __launch_bounds__(32) void mlp2_softmax_kernel(
    const float* __restrict__ Z,   // N x 128
    const float* __restrict__ W2,  // 128 x 16 row-major
    const float* __restrict__ b2,  // 16
    float* __restrict__ out,       // nb x 16
    int nb)
{
  __shared__ __align__(16) float sWT[16][132];  // sWT[n][k]
  const int lane = threadIdx.x;
  const int half = lane >> 4;
  const int lrow = lane & 15;
  const int m0   = blockIdx.x * 16;

  for (int i = lane; i < 2048; i += 32) {
    int n = i & 15, k = i >> 4;
    sWT[n][k] = W2[k * 16 + n];
  }
  __syncthreads();

  v8f acc = {0.f, 0.f, 0.f, 0.f, 0.f, 0.f, 0.f, 0.f};
#pragma unroll
  for (int k0 = 0; k0 < 128; k0 += 4) {
    const int kk = k0 + 2 * half;
    v2f a = *(const v2f*)&Z[(size_t)(m0 + lrow) * 128 + kk];
    v2f b = *(const v2f*)&sWT[lrow][kk];
    acc = WMMA_F32(a, b, acc);
  }

  const float bb = b2[lrow];
#pragma unroll
  for (int j = 0; j < 8; ++j) {
    float v = acc[j] + bb;
    float mx = v;
#pragma unroll
    for (int m = 1; m < 16; m <<= 1) mx = fmaxf(mx, __shfl_xor(mx, m, 32));
    float e = __expf(v - mx);
    float s = e;
#pragma unroll
    for (int m = 1; m < 16; m <<= 1) s += __shfl_xor(s, m, 32);
    const int r = m0 + j + 8 * half;
    if (r < nb) out[(size_t)r * 16 + lrow] = e / s;
  }
}

// ---------------------------------------------------------------------------
// host launcher
// ---------------------------------------------------------------------------
extern "C" void kernel_launch(void* const* d_in, const int* in_sizes, int n_in,
                              void* d_out, int out_size, void* d_ws, size_t ws_size,
                              hipStream_t stream)
{
  const float* x   = (const float*)d_in[0];
  const int*   src = (const int*)d_in[1];
  const int*   dst = (const int*)d_in[2];
  // d_in[3] = batch_size scalar (device); rows of output derived from out_size
  const float* W1  = (const float*)d_in[4];
  const float* b1  = (const float*)d_in[5];
  const float* W2  = (const float*)d_in[6];
  const float* b2  = (const float*)d_in[7];
  const float* W3  = (const float*)d_in[8];
  const float* b3  = (const float*)d_in[9];
  const float* Wm1 = (const float*)d_in[10];
  const float* bm1 = (const float*)d_in[11];
  const float* Wm2 = (const float*)d_in[12];
  const float* bm2 = (const float*)d_in[13];
  float* out = (float*)d_out;

  const int n  = in_sizes[0] / 128;  // 100000
  const int e  = in_sizes[1];        // 1600000
  const int nb = out_size / 16;

  // workspace carve-out (256B aligned slabs)
  char* ws = (char*)d_ws;
  const size_t szN  = (((size_t)n * 4) + 255) & ~(size_t)255;
  const size_t szNH = (((size_t)n * 128 * 4) + 255) & ~(size_t)255;
  float* dis  = (float*)(ws);
  float* Htmp = (float*)(ws + szN);               // GEMM result / later Z
  float* AGG  = (float*)(ws + szN + 1 * szNH);    // aggregation accumulator
  float* h1   = (float*)(ws + szN + 2 * szNH);
  float* h2   = (float*)(ws + szN + 3 * szNH);
  float* h3   = (float*)(ws + szN + 4 * szNH);

  // 1) symmetric normalization dis = rsqrt(deg) with self-loops
  fill_deg_kernel<<<(n + 255) / 256, 256, 0, stream>>>(dis, n);
  count_deg_kernel<<<(e + 255) / 256, 256, 0, stream>>>(dst, dis, e);
  rsqrt_kernel<<<(n + 255) / 256, 256, 0, stream>>>(dis, n);

  const int mtiles = n / 16;                      // 6250 (N divisible by 16)
  const long nthr32  = (long)n * 32;
  const int  gN32    = (int)((nthr32 + 255) / 256);
  const int  gEdge   = (e + 7) / 8;               // 8 edges per 256-thr block

  const float* Wl[3] = {W1, W2, W3};
  const float* bl[3] = {b1, b2, b3};
  float* hl[3] = {h1, h2, h3};
  const float* cur = x;
  for (int l = 0; l < 3; ++l) {
    gemm_wmma_kernel<1, false, false><<<mtiles, 128, 0, stream>>>(
        cur, nullptr, nullptr, nullptr, Wl[l], nullptr, Htmp);
    self_init_kernel<<<gN32, 256, 0, stream>>>(Htmp, dis, AGG, n);
    edge_agg_kernel<<<gEdge, 256, 0, stream>>>(Htmp, src, dst, dis, AGG, e);
    bias_relu_kernel<<<gN32, 256, 0, stream>>>(AGG, bl[l], hl[l], n);
    cur = hl[l];
  }

  // 2) jumping-knowledge MLP1: Z = relu([x|h1|h2|h3] @ Wm1 + bm1), K=512
  gemm_wmma_kernel<4, true, true><<<mtiles, 128, 0, stream>>>(
      x, h1, h2, h3, Wm1, bm1, Htmp /* Z */);

  // 3) MLP2 + softmax
  mlp2_softmax_kernel<<<mtiles, 32, 0, stream>>>(Htmp, Wm2, bm2, out, nb);
}